// Evolve_20701742367156
// MI455X (gfx1250) — compile-verified
//
#include <hip/hip_runtime.h>
#include <hip/hip_bf16.h>
#include <math.h>

// Problem constants from the reference
#define BB 8
#define NN 4096
#define EE 131072
#define FF 512
#define TT 3

typedef __attribute__((ext_vector_type(2))) float v2f;
typedef __attribute__((ext_vector_type(8))) float v8f;

// ---------------------------------------------------------------------------
// p_hat = pool_p / ||pool_p||
// ---------------------------------------------------------------------------
__global__ __launch_bounds__(256) void phat_kernel(const float* __restrict__ p,
                                                   float* __restrict__ p_hat) {
    __shared__ float red[256];
    float ss = 0.f;
    for (int i = threadIdx.x; i < FF; i += 256) ss += p[i] * p[i];
    red[threadIdx.x] = ss;
    __syncthreads();
    for (int s = 128; s > 0; s >>= 1) {
        if (threadIdx.x < s) red[threadIdx.x] += red[threadIdx.x + s];
        __syncthreads();
    }
    float inv = rsqrtf(red[0]);
    for (int i = threadIdx.x; i < FF; i += 256) p_hat[i] = p[i] * inv;
}

// ---------------------------------------------------------------------------
// score[n] = dot(x[n,:], p_hat)   (one wave32 per row)
// ---------------------------------------------------------------------------
__global__ __launch_bounds__(256) void score_kernel(const float* __restrict__ x,
                                                    const float* __restrict__ p_hat,
                                                    float* __restrict__ score) {
    int wave = (int)((blockIdx.x * blockDim.x + threadIdx.x) >> 5);
    int lane = threadIdx.x & 31;
    if (wave >= NN) return;
    const float* row = x + (size_t)wave * FF;
    float sum = 0.f;
    for (int f = lane; f < FF; f += 32) sum += row[f] * p_hat[f];
    for (int off = 16; off > 0; off >>= 1) sum += __shfl_xor(sum, off, 32);
    if (lane == 0) score[wave] = sum;
}

// ---------------------------------------------------------------------------
// Exact top-k (k=512) over 4096 scores: single-workgroup bitonic sort in LDS.
// ---------------------------------------------------------------------------
__global__ __launch_bounds__(512) void topk_kernel(const float* __restrict__ score,
                                                   float* __restrict__ topv,
                                                   int* __restrict__ topi) {
    __shared__ float s[NN];
    __shared__ int si[NN];
    for (int i = threadIdx.x; i < NN; i += 512) { s[i] = score[i]; si[i] = i; }
    __syncthreads();
    for (int k = 2; k <= NN; k <<= 1) {
        for (int j = k >> 1; j > 0; j >>= 1) {
            for (int i = threadIdx.x; i < NN; i += 512) {
                int ixj = i ^ j;
                if (ixj > i) {
                    bool desc = ((i & k) == 0);
                    float a = s[i], b = s[ixj];
                    bool sw = desc ? (a < b) : (a > b);
                    if (sw) {
                        s[i] = b; s[ixj] = a;
                        int t = si[i]; si[i] = si[ixj]; si[ixj] = t;
                    }
                }
            }
            __syncthreads();
        }
    }
    for (int i = threadIdx.x; i < FF; i += 512) { topv[i] = s[i]; topi[i] = si[i]; }
}

// ---------------------------------------------------------------------------
// x_tilde[i,f] = x[idx[i],f] * tanh(vals[i])
// ---------------------------------------------------------------------------
__global__ __launch_bounds__(256) void gather_kernel(const float* __restrict__ x,
                                                     const int* __restrict__ idx,
                                                     const float* __restrict__ vals,
                                                     float* __restrict__ xt) {
    int t = (int)(blockIdx.x * blockDim.x + threadIdx.x);   // FF*FF threads
    int r = t / FF, c = t % FF;
    xt[t] = x[(size_t)idx[r] * FF + c] * tanhf(vals[r]);
}

// ---------------------------------------------------------------------------
// FP32 WMMA GEMM:  C[M,N] = A[M,K] * op(B) (+bias[col])
//   TRANS_B=false : B is [K,N] row-major, staged as Bs[k][n]
//   TRANS_B=true  : B is [N,K] row-major (C = A*B^T), staged untransposed as
//                   Bs[n][k]; B-fragments are read transposed from LDS
//                   (stride-36 rows are bank-conflict-free across 16 lanes).
// Tile staging uses GLOBAL_LOAD_ASYNC_TO_LDS_B128 (ASYNCcnt) — no VGPR
// round-trip — then V_WMMA_F32_16X16X4_F32 on LDS-resident fragments.
// ---------------------------------------------------------------------------
#define GBM 64
#define GBN 128
#define GBK 32
#define LDA  36   // As row stride (floats): 16B-aligned, conflict-free frags
#define LDB 132   // Bs row stride, non-transposed [k][n] layout
#define LDBT 36   // Bs row stride, transposed-read [n][k] layout

__device__ __forceinline__ void async_ld_b128(unsigned lds_addr,
                                              const void* base,
                                              unsigned byte_off) {
    asm volatile("global_load_async_to_lds_b128 %0, %1, %2 offset:0"
                 :: "v"(lds_addr), "v"(byte_off), "s"(base)
                 : "memory");
}
__device__ __forceinline__ void wait_async0() {
    asm volatile("s_wait_asynccnt 0x0" ::: "memory");
}

template <bool TRANS_B>
__global__ __launch_bounds__(256) void gemm_f32_wmma(const float* __restrict__ A,
                                                     const float* __restrict__ B,
                                                     const float* __restrict__ bias,
                                                     float* __restrict__ C,
                                                     int M, int N, int K) {
    __shared__ __align__(16) float As[GBM * LDA];
    __shared__ __align__(16) float Bs[TRANS_B ? (GBN * LDBT) : (GBK * LDB)];

    const int tid  = threadIdx.x;
    const int lane = tid & 31;
    const int wid  = tid >> 5;       // 0..7
    const int wm   = wid >> 2;       // 0..1  (wave row)
    const int wn   = wid & 3;        // 0..3  (wave col)
    const int half = lane >> 4;      // ISA: lanes 16..31 carry K+2 / M+8
    const int l16  = lane & 15;

    const int block_n = blockIdx.x * GBN;
    const int block_m = blockIdx.y * GBM;

    const unsigned asbase = (unsigned)(uintptr_t)(&As[0]);
    const unsigned bsbase = (unsigned)(uintptr_t)(&Bs[0]);

    v8f acc[2][2] = {};

    for (int k0 = 0; k0 < K; k0 += GBK) {
        // ---- async-stage A tile 64x32 ----
        {
            int r  = tid >> 3;             // 0..31
            int c4 = (tid & 7) * 4;        // 0..28
            async_ld_b128(asbase + (unsigned)((r * LDA + c4) * 4), A,
                          (unsigned)(((block_m + r) * K + k0 + c4) * 4));
            async_ld_b128(asbase + (unsigned)(((r + 32) * LDA + c4) * 4), A,
                          (unsigned)(((block_m + r + 32) * K + k0 + c4) * 4));
            if (k0 + GBK < K)
                __builtin_prefetch(A + (size_t)(block_m + r) * K + k0 + GBK + c4, 0, 1);
        }
        // ---- async-stage B tile ----
        if (!TRANS_B) {
            int r  = tid >> 5;             // 0..7
            int c4 = (tid & 31) * 4;       // 0..124
            for (int it = 0; it < 4; ++it)
                async_ld_b128(bsbase + (unsigned)(((r + it * 8) * LDB + c4) * 4), B,
                              (unsigned)(((k0 + r + it * 8) * N + block_n + c4) * 4));
            if (k0 + GBK < K)
                __builtin_prefetch(B + (size_t)(k0 + GBK + r) * N + block_n + c4, 0, 1);
        } else {
            int nb = tid >> 3;             // 0..31
            int k4 = (tid & 7) * 4;        // 0..28
            for (int it = 0; it < 4; ++it) {
                int nn = nb + it * 32;
                async_ld_b128(bsbase + (unsigned)((nn * LDBT + k4) * 4), B,
                              (unsigned)(((block_n + nn) * K + k0 + k4) * 4));
            }
            if (k0 + GBK < K)
                __builtin_prefetch(B + (size_t)(block_n + nb) * K + k0 + GBK + k4, 0, 1);
        }
        wait_async0();
        __syncthreads();

        // ---- 8 x (K=4) WMMA slices ----
        for (int kk = 0; kk < GBK; kk += 4) {
            const int kidx = kk + 2 * half;
            v2f af[2], bf[2];
            for (int i = 0; i < 2; ++i) {
                int row = wm * 32 + i * 16 + l16;
                af[i].x = As[row * LDA + kidx];
                af[i].y = As[row * LDA + kidx + 1];
            }
            for (int j = 0; j < 2; ++j) {
                int col = wn * 32 + j * 16 + l16;
                if (TRANS_B) {
                    bf[j].x = Bs[col * LDBT + kidx];
                    bf[j].y = Bs[col * LDBT + kidx + 1];
                } else {
                    bf[j].x = Bs[kidx * LDB + col];
                    bf[j].y = Bs[(kidx + 1) * LDB + col];
                }
            }
            for (int i = 0; i < 2; ++i)
                for (int j = 0; j < 2; ++j)
                    acc[i][j] = __builtin_amdgcn_wmma_f32_16x16x4_f32(
                        false, af[i], false, bf[j], (short)0, acc[i][j], false, false);
        }
        __syncthreads();
    }

    // ---- epilogue: C/D layout -> row = base + half*8 + r, col = base + l16 ----
    for (int i = 0; i < 2; ++i)
        for (int j = 0; j < 2; ++j) {
            int col = block_n + wn * 32 + j * 16 + l16;
            float bv = bias ? bias[col] : 0.0f;
            int row0 = block_m + wm * 32 + i * 16 + half * 8;
            for (int r = 0; r < 8; ++r)
                C[(size_t)(row0 + r) * N + col] = acc[i][j][r] + bv;
        }
}

// ---------------------------------------------------------------------------
// GRU gate fusion: h <- (1-z)*n + z*h  from gi,gh (each [F, 3F]) and old h.
// ---------------------------------------------------------------------------
__global__ __launch_bounds__(256) void gru_gate_kernel(const float* __restrict__ gi,
                                                       const float* __restrict__ gh,
                                                       float* __restrict__ h) {
    int t = (int)(blockIdx.x * blockDim.x + threadIdx.x);   // FF*FF
    int m = t / FF, c = t % FF;
    const float* gim = gi + (size_t)m * 3 * FF;
    const float* ghm = gh + (size_t)m * 3 * FF;
    float ir = gim[c],           hr = ghm[c];
    float iz = gim[FF + c],      hz = ghm[FF + c];
    float inn = gim[2 * FF + c], hn = ghm[2 * FF + c];
    float r = 1.f / (1.f + expf(-(ir + hr)));
    float z = 1.f / (1.f + expf(-(iz + hz)));
    float n = tanhf(inn + r * hn);
    float hv = h[t];
    h[t] = (1.f - z) * n + z * hv;
}

// ---------------------------------------------------------------------------
// GCN normalization + scatter aggregation
// ---------------------------------------------------------------------------
__global__ __launch_bounds__(256) void deg_init_kernel(float* __restrict__ deg) {
    int i = (int)(blockIdx.x * blockDim.x + threadIdx.x);
    if (i < NN) deg[i] = 1.0f;   // self-loop weight
}

__global__ __launch_bounds__(256) void edge_deg_kernel(const int* __restrict__ ei,
                                                       const float* __restrict__ ea,
                                                       float* __restrict__ deg) {
    int e = (int)(blockIdx.x * blockDim.x + threadIdx.x);
    if (e >= EE) return;
    atomicAdd(&deg[ei[e]], ea[(size_t)e * TT]);   // row = ei[0*E+e], w = ea[e,0]
}

__global__ __launch_bounds__(256) void dinv_kernel(const float* __restrict__ deg,
                                                   float* __restrict__ dinv) {
    int i = (int)(blockIdx.x * blockDim.x + threadIdx.x);
    if (i >= NN) return;
    float d = deg[i];
    dinv[i] = d > 0.f ? rsqrtf(d) : 0.f;
}

// out[i,f] = dinv[i]^2 * xw[i,f] + bias[f]   (self-loop term, bias fused)
__global__ __launch_bounds__(256) void agg_init_kernel(const float* __restrict__ xw,
                                                       const float* __restrict__ dinv,
                                                       const float* __restrict__ bias,
                                                       float* __restrict__ out) {
    int t = (int)(blockIdx.x * blockDim.x + threadIdx.x);   // NN*FF
    int i = t / FF, f = t % FF;
    float di = dinv[i];
    out[t] = di * di * xw[t] + bias[f];
}

// One wave32 per edge: out[row,:] += dinv[row]*w*dinv[col] * xw[col,:]
// (xw is 8 MB -> L2-resident on the 192 MB MI455X L2; atomics execute at L2)
__global__ __launch_bounds__(256) void edge_agg_kernel(const int* __restrict__ ei,
                                                       const float* __restrict__ ea,
                                                       const float* __restrict__ dinv,
                                                       const float* __restrict__ xw,
                                                       float* __restrict__ out) {
    int gt = (int)(blockIdx.x * blockDim.x + threadIdx.x);
    int e = gt >> 5, lane = gt & 31;
    if (e >= EE) return;
    int r = ei[e], c = ei[EE + e];
    float coef = dinv[r] * ea[(size_t)e * TT] * dinv[c];
    const float* src = xw + (size_t)c * FF;
    float* dst = out + (size_t)r * FF;
    for (int f = lane; f < FF; f += 32)
        atomicAdd(&dst[f], coef * src[f]);
}

// ---------------------------------------------------------------------------
// Driver
// ---------------------------------------------------------------------------
extern "C" void kernel_launch(void* const* d_in, const int* in_sizes, int n_in,
                              void* d_out, int out_size, void* d_ws, size_t ws_size,
                              hipStream_t stream) {
    const float* x        = (const float*)d_in[0];   // (B,N,F)
    const int*   edge_idx = (const int*)  d_in[1];   // (B,2,E)
    const float* edge_att = (const float*)d_in[2];   // (B,E,T)
    const float* pool_p   = (const float*)d_in[3];   // (F,)
    const float* c1_w     = (const float*)d_in[4];   // (F,F)
    const float* c1_b     = (const float*)d_in[5];
    const float* c2_w     = (const float*)d_in[6];
    const float* c2_b     = (const float*)d_in[7];
    const float* g1_wih   = (const float*)d_in[8];   // (3F,F)
    const float* g1_whh   = (const float*)d_in[9];
    const float* g1_bih   = (const float*)d_in[10];
    const float* g1_bhh   = (const float*)d_in[11];
    const float* g2_wih   = (const float*)d_in[12];
    const float* g2_whh   = (const float*)d_in[13];
    const float* g2_bih   = (const float*)d_in[14];
    const float* g2_bhh   = (const float*)d_in[15];
    float* out = (float*)d_out;

    // workspace carve-up (floats)
    float* ws    = (float*)d_ws;
    float* p_hat = ws;                    // F
    float* score = p_hat + FF;            // N
    float* topv  = score + NN;            // F
    int*   topi  = (int*)(topv + FF);     // F ints
    float* xtld  = topv + 2 * FF;         // F*F
    float* W1    = xtld + FF * FF;        // F*F (carry)
    float* W2    = W1 + FF * FF;          // F*F (carry)
    float* gi    = W2 + FF * FF;          // F*3F
    float* gh    = gi + FF * 3 * FF;      // F*3F
    float* xw    = gh + FF * 3 * FF;      // N*F
    float* h1    = xw + (size_t)NN * FF;  // N*F
    float* deg   = h1 + (size_t)NN * FF;  // N
    float* dinv  = deg + NN;              // N

    // carry init: W1 <- conv1_weight, W2 <- conv2_weight
    hipMemcpyAsync(W1, c1_w, (size_t)FF * FF * sizeof(float), hipMemcpyDeviceToDevice, stream);
    hipMemcpyAsync(W2, c2_w, (size_t)FF * FF * sizeof(float), hipMemcpyDeviceToDevice, stream);

    phat_kernel<<<1, 256, 0, stream>>>(pool_p, p_hat);

    const dim3 gemm_blk(256);
    const dim3 grid_gru(3 * FF / GBN, FF / GBM);   // 12 x 8
    const dim3 grid_big(FF / GBN, NN / GBM);       // 4 x 64

    for (int t = 0; t < BB; ++t) {
        const float* xt   = x + (size_t)t * NN * FF;
        const int*   ei   = edge_idx + (size_t)t * 2 * EE;
        const float* ea   = edge_att + (size_t)t * EE * TT;
        float*       outt = out + (size_t)t * NN * FF;

        // pooling: score, top-k, gather
        score_kernel<<<NN / 8, 256, 0, stream>>>(xt, p_hat, score);
        topk_kernel<<<1, 512, 0, stream>>>(score, topv, topi);
        gather_kernel<<<FF * FF / 256, 256, 0, stream>>>(xt, topi, topv, xtld);

        // GRU1: W1 <- gru(x_tilde, W1)
        gemm_f32_wmma<true><<<grid_gru, gemm_blk, 0, stream>>>(xtld, g1_wih, g1_bih, gi, FF, 3 * FF, FF);
        gemm_f32_wmma<true><<<grid_gru, gemm_blk, 0, stream>>>(W1,   g1_whh, g1_bhh, gh, FF, 3 * FF, FF);
        gru_gate_kernel<<<FF * FF / 256, 256, 0, stream>>>(gi, gh, W1);

        // GCN1: h1 = gcn(xt, W1)
        gemm_f32_wmma<false><<<grid_big, gemm_blk, 0, stream>>>(xt, W1, nullptr, xw, NN, FF, FF);
        deg_init_kernel<<<NN / 256, 256, 0, stream>>>(deg);
        edge_deg_kernel<<<EE / 256, 256, 0, stream>>>(ei, ea, deg);
        dinv_kernel<<<NN / 256, 256, 0, stream>>>(deg, dinv);
        agg_init_kernel<<<NN * FF / 256, 256, 0, stream>>>(xw, dinv, c1_b, h1);
        edge_agg_kernel<<<EE * 32 / 256, 256, 0, stream>>>(ei, ea, dinv, xw, h1);

        // GRU2: W2 <- gru(W2, W2)
        gemm_f32_wmma<true><<<grid_gru, gemm_blk, 0, stream>>>(W2, g2_wih, g2_bih, gi, FF, 3 * FF, FF);
        gemm_f32_wmma<true><<<grid_gru, gemm_blk, 0, stream>>>(W2, g2_whh, g2_bhh, gh, FF, 3 * FF, FF);
        gru_gate_kernel<<<FF * FF / 256, 256, 0, stream>>>(gi, gh, W2);

        // GCN2: out_t = gcn(h1, W2)
        gemm_f32_wmma<false><<<grid_big, gemm_blk, 0, stream>>>(h1, W2, nullptr, xw, NN, FF, FF);
        agg_init_kernel<<<NN * FF / 256, 256, 0, stream>>>(xw, dinv, c2_b, outt);
        edge_agg_kernel<<<EE * 32 / 256, 256, 0, stream>>>(ei, ea, dinv, xw, outt);
    }
    (void)in_sizes; (void)n_in; (void)out_size; (void)ws_size;
}